// attentionb_5892695130374
// MI455X (gfx1250) — compile-verified
//
#include <hip/hip_runtime.h>
#include <math.h>

// ---------------------------------------------------------------------------
// attentionb for MI455X (gfx1250, wave32).
// Pipeline:
//   K1 rmsnorm3      : xqn = rms(x)*gq, xkvn = rms(x)*gkv, ptn = rms(pt)*gc
//   K2 wmma_gemm_bias: Q = xqn@Wq+bq ; KV = xkvn@Wkv+bkv ; C = ptn@Wc+bc
//   K3 cumsum        : P[t+1] = P[t] + C[t]  (per (head,dim) column), P[0]=0
//   K4 augment       : Qa_i = [q_i*(scale+PE_i), q_i], Ka_j = [k_j, -k_j*PS_j]
//   K5 attn          : causal flash attention per head with 128-dim Q~K~ scores
//   K6 wmma_gemm_bias: out = wv@Wout + bout
// All matmuls via V_WMMA_F32_16X16X4_F32 (full fp32, matches fp32 reference).
// ---------------------------------------------------------------------------

typedef __attribute__((ext_vector_type(2))) float v2f;
typedef __attribute__((ext_vector_type(8))) float v8f;

#define L   1024
#define DM  1024
#define EPSR 1.1920929e-07f

static __device__ __forceinline__ v8f wmma_f32(v2f a, v2f b, v8f c) {
  // 8 args: (neg_a, A, neg_b, B, c_mod, C, reuse_a, reuse_b)
  return __builtin_amdgcn_wmma_f32_16x16x4_f32(false, a, false, b, (short)0, c,
                                               false, false);
}

// ---------------------------------------------------------------- K1: rmsnorm
__global__ void __launch_bounds__(256)
rmsnorm3_kernel(const float* __restrict__ x, const float* __restrict__ pt,
                const float* __restrict__ gq, const float* __restrict__ gkv,
                const float* __restrict__ gc,
                float* __restrict__ xqn, float* __restrict__ xkvn,
                float* __restrict__ ptn) {
  __shared__ float red[256];
  const int row = blockIdx.x;
  const int tid = threadIdx.x;

  float4 xv = ((const float4*)(x + (size_t)row * DM))[tid];
  red[tid] = xv.x * xv.x + xv.y * xv.y + xv.z * xv.z + xv.w * xv.w;
  __syncthreads();
  for (int s = 128; s > 0; s >>= 1) {
    if (tid < s) red[tid] += red[tid + s];
    __syncthreads();
  }
  float rn = rsqrtf(red[0] * (1.0f / DM) + EPSR);
  __syncthreads();

  float4 gqv = ((const float4*)gq)[tid];
  float4 gkvv = ((const float4*)gkv)[tid];
  float4 oq, okv;
  oq.x = xv.x * rn * gqv.x;  oq.y = xv.y * rn * gqv.y;
  oq.z = xv.z * rn * gqv.z;  oq.w = xv.w * rn * gqv.w;
  okv.x = xv.x * rn * gkvv.x; okv.y = xv.y * rn * gkvv.y;
  okv.z = xv.z * rn * gkvv.z; okv.w = xv.w * rn * gkvv.w;
  ((float4*)(xqn + (size_t)row * DM))[tid] = oq;
  ((float4*)(xkvn + (size_t)row * DM))[tid] = okv;

  float4 pv = ((const float4*)(pt + (size_t)row * DM))[tid];
  red[tid] = pv.x * pv.x + pv.y * pv.y + pv.z * pv.z + pv.w * pv.w;
  __syncthreads();
  for (int s = 128; s > 0; s >>= 1) {
    if (tid < s) red[tid] += red[tid + s];
    __syncthreads();
  }
  float rnp = rsqrtf(red[0] * (1.0f / DM) + EPSR);
  float4 gcv = ((const float4*)gc)[tid];
  float4 oc;
  oc.x = pv.x * rnp * gcv.x; oc.y = pv.y * rnp * gcv.y;
  oc.z = pv.z * rnp * gcv.z; oc.w = pv.w * rnp * gcv.w;
  ((float4*)(ptn + (size_t)row * DM))[tid] = oc;
}

// --------------------------------------------------- K2/K6: fp32 WMMA GEMM+b
// C[M=1024, N] = A[M,1024] @ W[1024,N] + bias.  64x64 block tile, 8 waves,
// each wave owns a 16x32 output (2 accumulators). LDS-staged, 16-wide K steps.
__global__ void __launch_bounds__(256)
wmma_gemm_bias(const float* __restrict__ A, const float* __restrict__ Wm,
               const float* __restrict__ bias, float* __restrict__ Cout,
               int N) {
  __shared__ float As[16 * 65];  // [k][m], padded pitch -> conflict-free
  __shared__ float Bs[16 * 64];  // [k][n]
  const int tid = threadIdx.x;
  const int lane = tid & 31, wid = tid >> 5;
  const int hhi = lane >> 4, l16 = lane & 15;
  const int m0 = blockIdx.x * 64, n0 = blockIdx.y * 64;
  const int wm = wid & 3, wn = wid >> 2;
  const int mloc = wm * 16 + l16;
  const int nloc0 = wn * 32 + l16;
  const int nloc1 = nloc0 + 16;

  v8f acc0 = {}; v8f acc1 = {};

  for (int k0 = 0; k0 < DM; k0 += 16) {
#pragma unroll
    for (int it = 0; it < 4; ++it) {
      int idx = tid + it * 256;
      int ak = idx & 15, am = idx >> 4;
      As[ak * 65 + am] = A[(size_t)(m0 + am) * DM + k0 + ak];
      int bn = idx & 63, bk = idx >> 6;
      Bs[bk * 64 + bn] = Wm[(size_t)(k0 + bk) * N + n0 + bn];
    }
    __syncthreads();
    if (k0 + 16 < DM) {  // -> global_prefetch_b8
      __builtin_prefetch(&Wm[(size_t)(k0 + 16 + (tid >> 6)) * N + n0 + (tid & 63)], 0, 1);
      __builtin_prefetch(&A[(size_t)(m0 + (tid >> 2)) * DM + k0 + 16], 0, 1);
    }
#pragma unroll
    for (int kk = 0; kk < 4; ++kk) {
      const int kb = kk * 4 + 2 * hhi;
      v2f a;  a.x = As[kb * 65 + mloc];        a.y = As[(kb + 1) * 65 + mloc];
      v2f b0; b0.x = Bs[kb * 64 + nloc0];      b0.y = Bs[(kb + 1) * 64 + nloc0];
      v2f b1; b1.x = Bs[kb * 64 + nloc1];      b1.y = Bs[(kb + 1) * 64 + nloc1];
      acc0 = wmma_f32(a, b0, acc0);
      acc1 = wmma_f32(a, b1, acc1);
    }
    __syncthreads();
  }

  const int rbase = m0 + wm * 16 + hhi * 8;
  const float bb0 = bias[n0 + nloc0];
  const float bb1 = bias[n0 + nloc1];
#pragma unroll
  for (int v = 0; v < 8; ++v) {
    const int r = rbase + v;
    Cout[(size_t)r * N + n0 + nloc0] = acc0[v] + bb0;
    Cout[(size_t)r * N + n0 + nloc1] = acc1[v] + bb1;
  }
}

// ------------------------------------------------------------- K3: prefix sum
__global__ void cumsum_kernel(const float* __restrict__ C, float* __restrict__ P) {
  const int col = blockIdx.x * blockDim.x + threadIdx.x;  // 1024 columns
  float run = 0.f;
  P[col] = 0.f;
  for (int t = 0; t < L; ++t) {
    run += C[(size_t)t * DM + col];
    P[(size_t)(t + 1) * DM + col] = run;
  }
}

// ----------------------------------------------------------- K4: augmentation
__global__ void augment_kernel(const float* __restrict__ Q,
                               const float* __restrict__ KV,
                               const float* __restrict__ P,
                               const int* __restrict__ cw,
                               float* __restrict__ Qa, float* __restrict__ Ka) {
  const int i = blockIdx.x;
  const int w = cw[0];
  const float scale = 0.125f;  // d^-0.5, d=64
  int E = i + w; if (E > L) E = L;
  int S = i - w; if (S < 0) S = 0;
  for (int c = threadIdx.x; c < DM; c += blockDim.x) {
    const int h = c >> 6, dd = c & 63;
    const float pe = P[(size_t)E * DM + c];
    const float ps = P[(size_t)S * DM + c];
    const float qv = Q[(size_t)i * DM + c];
    const float kv = KV[(size_t)i * 2048 + c];  // k = first 1024 cols of KV
    const int base = i * 2048 + h * 128 + dd;   // per-head 128-wide layout
    Qa[base]      = qv * (scale + pe);
    Qa[base + 64] = qv;
    Ka[base]      = kv;
    Ka[base + 64] = -kv * ps;
  }
}

// -------------------------------------------------- K5: causal flash attention
// Block = 4 waves = 64 query rows of one head. Each wave owns 16 rows.
// Scores via Q~(16x128) @ K~^T using 32 fp32 WMMAs; online softmax; P@V via 16.
__global__ void __launch_bounds__(128)
attn_kernel(const float* __restrict__ Qa, const float* __restrict__ Ka,
            const float* __restrict__ KV, float* __restrict__ wv) {
  __shared__ float Qs[4 * 16 * 130];  // per-wave Q~ tile [m][k], pitch 130
  __shared__ float Ks[16 * 130];      // K~ tile [j][k]
  __shared__ float Vs[16 * 64];       // V tile [j][n]
  __shared__ float Pls[4 * 16 * 18];  // per-wave P tile [m][j], pitch 18

  const int tid = threadIdx.x;
  const int lane = tid & 31, wid = tid >> 5;
  const int hhi = lane >> 4, l16 = lane & 15;
  const int h = blockIdx.y;
  const int hoff = h * 128;
  const int qrow0 = blockIdx.x * 64 + wid * 16;
  const int qtile = qrow0 >> 4;
  const int jt_max = blockIdx.x * 4 + 3;

  float* myQ = &Qs[wid * 16 * 130];
  float* myP = &Pls[wid * 16 * 18];

  // Stage this wave's 16x128 Q~ tile (wave-private, no block barrier needed).
  for (int idx = lane; idx < 16 * 128; idx += 32) {
    const int r = idx >> 7, kc = idx & 127;
    myQ[r * 130 + kc] = Qa[(size_t)(qrow0 + r) * 2048 + hoff + kc];
  }

  v8f o0 = {}, o1 = {}, o2 = {}, o3 = {};
  float m8[8], l8[8];
#pragma unroll
  for (int v = 0; v < 8; ++v) { m8[v] = -3.0e38f; l8[v] = 0.f; }

  for (int jt = 0; jt <= jt_max; ++jt) {
    // Cooperative stage of K~ (16x128) and V (16x64) for key tile jt.
    for (int idx = tid; idx < 16 * 128; idx += 128) {
      const int r = idx >> 7, kc = idx & 127;
      Ks[r * 130 + kc] = Ka[(size_t)(jt * 16 + r) * 2048 + hoff + kc];
    }
    for (int idx = tid; idx < 16 * 64; idx += 128) {
      const int r = idx >> 6, n = idx & 63;
      Vs[r * 64 + n] = KV[(size_t)(jt * 16 + r) * 2048 + 1024 + h * 64 + n];
    }
    __syncthreads();

    if (jt <= qtile) {  // wave-uniform: EXEC stays all-ones for WMMA
      // S(16x16) = Q~ @ K~^T over K=128
      v8f s = {};
#pragma unroll 8
      for (int kk = 0; kk < 32; ++kk) {
        const int kb = kk * 4 + 2 * hhi;
        v2f a; a.x = myQ[l16 * 130 + kb]; a.y = myQ[l16 * 130 + kb + 1];
        v2f b; b.x = Ks[l16 * 130 + kb];  b.y = Ks[l16 * 130 + kb + 1];
        s = wmma_f32(a, b, s);
      }

      const bool diag = (jt == qtile);
#pragma unroll
      for (int v = 0; v < 8; ++v) {
        const int row = v + hhi * 8;
        float sv = s[v];
        if (diag && (l16 > row)) sv = -1.0e30f;  // causal: exp -> exactly 0
        float mx = sv;
        for (int off = 8; off > 0; off >>= 1)
          mx = fmaxf(mx, __shfl_xor(mx, off, 16));
        const float mnew = fmaxf(m8[v], mx);
        const float alpha = __expf(m8[v] - mnew);
        const float p = __expf(sv - mnew);
        float rs = p;
        for (int off = 8; off > 0; off >>= 1) rs += __shfl_xor(rs, off, 16);
        l8[v] = l8[v] * alpha + rs;
        m8[v] = mnew;
        o0[v] *= alpha; o1[v] *= alpha; o2[v] *= alpha; o3[v] *= alpha;
        myP[row * 18 + l16] = p;  // C-layout -> A-layout via LDS
      }

      // O += P(16x16) @ V(16x64)
#pragma unroll
      for (int kk = 0; kk < 4; ++kk) {
        const int kb = kk * 4 + 2 * hhi;
        v2f a; a.x = myP[l16 * 18 + kb]; a.y = myP[l16 * 18 + kb + 1];
        v2f b0, b1, b2, b3;
        b0.x = Vs[kb * 64 + l16];      b0.y = Vs[(kb + 1) * 64 + l16];
        b1.x = Vs[kb * 64 + 16 + l16]; b1.y = Vs[(kb + 1) * 64 + 16 + l16];
        b2.x = Vs[kb * 64 + 32 + l16]; b2.y = Vs[(kb + 1) * 64 + 32 + l16];
        b3.x = Vs[kb * 64 + 48 + l16]; b3.y = Vs[(kb + 1) * 64 + 48 + l16];
        o0 = wmma_f32(a, b0, o0);
        o1 = wmma_f32(a, b1, o1);
        o2 = wmma_f32(a, b2, o2);
        o3 = wmma_f32(a, b3, o3);
      }
    }
    __syncthreads();  // before next tile overwrites Ks/Vs
  }

  const int cb = h * 64;
#pragma unroll
  for (int v = 0; v < 8; ++v) {
    const float inv = 1.0f / l8[v];
    const int r = qrow0 + v + hhi * 8;
    wv[(size_t)r * DM + cb + l16]      = o0[v] * inv;
    wv[(size_t)r * DM + cb + 16 + l16] = o1[v] * inv;
    wv[(size_t)r * DM + cb + 32 + l16] = o2[v] * inv;
    wv[(size_t)r * DM + cb + 48 + l16] = o3[v] * inv;
  }
}

// ---------------------------------------------------------------------------
extern "C" void kernel_launch(void* const* d_in, const int* in_sizes, int n_in,
                              void* d_out, int out_size, void* d_ws,
                              size_t ws_size, hipStream_t stream) {
  const float* x    = (const float*)d_in[0];
  const float* pt   = (const float*)d_in[1];
  // d_in[2] (mask) realized analytically: -1e9 additive causal mask gives
  // exactly-zero softmax weights in fp32, reproduced by the causal flash loop.
  const float* gq   = (const float*)d_in[3];
  const float* Wq   = (const float*)d_in[4];
  const float* bq   = (const float*)d_in[5];
  const float* gkv  = (const float*)d_in[6];
  const float* Wkv  = (const float*)d_in[7];
  const float* bkv  = (const float*)d_in[8];
  const float* gc   = (const float*)d_in[9];
  const float* Wc   = (const float*)d_in[10];
  const float* bc   = (const float*)d_in[11];
  const float* Wout = (const float*)d_in[12];
  const float* bout = (const float*)d_in[13];
  const int*   cw   = (const int*)d_in[14];
  float* out = (float*)d_out;

  float* ws = (float*)d_ws;
  const size_t R = 1024;  // floats per "row unit"
  float* xqn  = ws + 0 * R * 1024;      // 1024 x 1024
  float* xkvn = ws + 1 * R * 1024;      // 1024 x 1024
  float* ptn  = ws + 2 * R * 1024;      // 1024 x 1024
  float* Qb   = ws + 3 * R * 1024;      // 1024 x 1024
  float* KVb  = ws + 4 * R * 1024;      // 1024 x 2048
  float* Cb   = ws + 6 * R * 1024;      // 1024 x 1024
  float* Pb   = ws + 7 * R * 1024;      // 1025 x 1024 (padded to 1088 rows)
  float* Qab  = ws + 8256 * R;          // 1024 x 2048
  float* Kab  = ws + 10304 * R;         // 1024 x 2048
  float* wvb  = ws + 12352 * R;         // 1024 x 1024   (total ~52.3 MB)

  rmsnorm3_kernel<<<1024, 256, 0, stream>>>(x, pt, gq, gkv, gc, xqn, xkvn, ptn);
  wmma_gemm_bias<<<dim3(16, 16), 256, 0, stream>>>(xqn, Wq, bq, Qb, 1024);
  wmma_gemm_bias<<<dim3(16, 32), 256, 0, stream>>>(xkvn, Wkv, bkv, KVb, 2048);
  wmma_gemm_bias<<<dim3(16, 16), 256, 0, stream>>>(ptn, Wc, bc, Cb, 1024);
  cumsum_kernel<<<4, 256, 0, stream>>>(Cb, Pb);
  augment_kernel<<<1024, 256, 0, stream>>>(Qb, KVb, Pb, cw, Qab, Kab);
  attn_kernel<<<dim3(16, 16), 128, 0, stream>>>(Qab, Kab, KVb, wvb);
  wmma_gemm_bias<<<dim3(16, 16), 256, 0, stream>>>(wvb, Wout, bout, out, 1024);
}